// RTEFCOS_67216238183088
// MI455X (gfx1250) — compile-verified
//
#include <hip/hip_runtime.h>
#include <hip/hip_bf16.h>

#define NLEV        5
#define NUM_CLS     80
#define TOPK_N      1000
#define NCAT        4256
#define NBLK        67          /* ceil(4256/64) */
#define CONF_TH     0.05f
#define NMS_TH      0.6f
#define NBUF        8           /* async ring depth in decode */
#define TOT_POS     87296

#if defined(__HIP_DEVICE_COMPILE__) && defined(__gfx1250__)
#define GFX1250_ASYNC 1
#endif

// ---------------------------------------------------------------------------
// helpers
// ---------------------------------------------------------------------------
__device__ __forceinline__ float sigf(float x) { return 1.0f / (1.0f + __expf(-x)); }

// descending bitonic sort of n u64 keys in LDS; all threads of block participate
__device__ inline void bitonic_desc(unsigned long long* a, int n) {
    for (int k = 2; k <= n; k <<= 1) {
        for (int j = k >> 1; j > 0; j >>= 1) {
            __syncthreads();
            for (int t = threadIdx.x; t < n; t += blockDim.x) {
                int ixj = t ^ j;
                if (ixj > t) {
                    bool up = ((t & k) == 0);   // "up" segments sorted descending
                    unsigned long long x = a[t], y = a[ixj];
                    bool sw = up ? (x < y) : (x > y);
                    if (sw) { a[t] = y; a[ixj] = x; }
                }
            }
        }
    }
    __syncthreads();
}

// ---------------------------------------------------------------------------
// K1: per-level decode  (scores as orderable uint bits + argmax label)
//     class planes streamed through an 8-deep async global->LDS ring
// ---------------------------------------------------------------------------
__global__ void decode_kernel(const float* __restrict__ cls,
                              const float* __restrict__ ctn,
                              unsigned* __restrict__ keys,
                              int* __restrict__ labels,
                              int HW) {
    const int pos = blockIdx.x * 256 + threadIdx.x;   // HW is a multiple of 256
    float m = -1e30f;
    int   lbl = 0;

#ifdef GFX1250_ASYNC
    __shared__ float buf[NBUF][256];
    // prologue: fill the ring
    for (int c = 0; c < NBUF; ++c) {
        unsigned           la = (unsigned)(size_t)&buf[c][threadIdx.x];
        unsigned long long ga = (unsigned long long)(size_t)(cls + (size_t)c * HW + pos);
        asm volatile("global_load_async_to_lds_b32 %0, %1, off" :: "v"(la), "v"(ga) : "memory");
    }
    // steady state: consume channel c, refill slot with channel c+NBUF
    for (int c = 0; c < NUM_CLS - NBUF; ++c) {
        asm volatile("s_wait_asynccnt 7" ::: "memory");   // oldest of 8 done (per-wave)
        float v = buf[c & (NBUF - 1)][threadIdx.x];
        if (v > m) { m = v; lbl = c; }
        asm volatile("s_wait_dscnt 0" ::: "memory");      // ds read done before DMA reuses slot
        const int cn = c + NBUF;
        unsigned           la = (unsigned)(size_t)&buf[cn & (NBUF - 1)][threadIdx.x];
        unsigned long long ga = (unsigned long long)(size_t)(cls + (size_t)cn * HW + pos);
        asm volatile("global_load_async_to_lds_b32 %0, %1, off" :: "v"(la), "v"(ga) : "memory");
    }
    asm volatile("s_wait_asynccnt 0" ::: "memory");
    for (int c = NUM_CLS - NBUF; c < NUM_CLS; ++c) {
        float v = buf[c & (NBUF - 1)][threadIdx.x];
        if (v > m) { m = v; lbl = c; }
    }
#else
    for (int c = 0; c < NUM_CLS; ++c) {
        if (c + 4 < NUM_CLS) __builtin_prefetch(cls + (size_t)(c + 4) * HW + pos, 0, 1);
        float v = cls[(size_t)c * HW + pos];
        if (v > m) { m = v; lbl = c; }
    }
#endif

    const float t = ctn[pos];
    const float score = sqrtf(sigf(m) * sigf(t));   // sqrt(sig(max)*sig(ctn)); argmax(prob)=argmax(cls)
    keys[pos]   = __float_as_uint(score);           // score >= 0 -> bits are order-preserving
    labels[pos] = lbl;
}

// ---------------------------------------------------------------------------
// K2: exact per-level top-k (one workgroup): radix-select threshold, compact,
//     bitonic sort 2048 (key desc, idx asc), emit first TOPK_N indices.
// ---------------------------------------------------------------------------
__global__ void topk_kernel(const unsigned* __restrict__ keys, int P,
                            int* __restrict__ selOut) {
    if (P <= TOPK_N) {                       // reference skips top_k entirely
        for (int i = threadIdx.x; i < P; i += blockDim.x) selOut[i] = i;
        return;
    }
    __shared__ unsigned hist[256];
    __shared__ unsigned s_prefix, s_remaining;
    __shared__ unsigned long long comb[2048];
    __shared__ unsigned cnt;

    unsigned prefix = 0, remaining = TOPK_N;
    for (int shift = 24; shift >= 0; shift -= 8) {
        for (int b = threadIdx.x; b < 256; b += blockDim.x) hist[b] = 0;
        __syncthreads();
        const unsigned maskHigh = (shift == 24) ? 0u : (0xFFFFFFFFu << (shift + 8));
        for (int p = threadIdx.x; p < P; p += blockDim.x) {
            unsigned k = keys[p];
            if ((k & maskHigh) == (prefix & maskHigh))
                atomicAdd(&hist[(k >> shift) & 255], 1u);
        }
        __syncthreads();
        if (threadIdx.x == 0) {
            unsigned cum = 0; int bsel = 0;
            for (int b = 255; b >= 0; --b) {
                if (cum + hist[b] >= remaining) { bsel = b; break; }
                cum += hist[b];
            }
            s_prefix    = prefix | ((unsigned)bsel << shift);
            s_remaining = remaining - cum;
        }
        __syncthreads();
        prefix = s_prefix; remaining = s_remaining;
        __syncthreads();
    }
    // compact everything >= threshold key (count in [TOPK_N, TOPK_N + ties])
    for (int i = threadIdx.x; i < 2048; i += blockDim.x)
        comb[i] = (unsigned long long)(unsigned)(~0x7fffffff);   // key 0, huge idx -> sorts last
    if (threadIdx.x == 0) cnt = 0;
    __syncthreads();
    for (int p = threadIdx.x; p < P; p += blockDim.x) {
        unsigned k = keys[p];
        if (k >= prefix) {
            unsigned q = atomicAdd(&cnt, 1u);
            if (q < 2048) comb[q] = ((unsigned long long)k << 32) | (unsigned)(~p);
        }
    }
    __syncthreads();
    bitonic_desc(comb, 2048);                // key desc, then idx asc (via ~idx)
    for (int r = threadIdx.x; r < TOPK_N; r += blockDim.x)
        selOut[r] = (int)(~(unsigned)comb[r]);
}

// ---------------------------------------------------------------------------
// K3: gather selected candidates -> concatenated score/label/box/box+offset
// ---------------------------------------------------------------------------
__global__ void gather_kernel(const int* __restrict__ selIdx,
                              const unsigned* __restrict__ keys,
                              const int* __restrict__ labels,
                              const float* __restrict__ reg0, const float* __restrict__ reg1,
                              const float* __restrict__ reg2, const float* __restrict__ reg3,
                              const float* __restrict__ reg4,
                              const float* __restrict__ scales,
                              float* __restrict__ catScore, int* __restrict__ catLabel,
                              float* __restrict__ catBox, float* __restrict__ catBoxOff) {
    int s = blockIdx.x * blockDim.x + threadIdx.x;
    if (s >= NCAT) return;
    const int   Pl[5]   = {65536, 16384, 4096, 1024, 256};
    const int   Wl[5]   = {256, 128, 64, 32, 16};
    const int   SOff[5] = {0, 65536, 81920, 86016, 87040};
    const float Stl[5]  = {8.f, 16.f, 32.f, 64.f, 128.f};
    int l = s / 1000; if (l > 4) l = 4;
    const int idx = selIdx[s];
    const int P = Pl[l], W = Wl[l];
    const float* rg = (l == 0) ? reg0 : (l == 1) ? reg1 : (l == 2) ? reg2 : (l == 3) ? reg3 : reg4;
    const float sc = scales[l];
    const float st = Stl[l];
    const unsigned kb = keys[SOff[l] + idx];
    const int lab = labels[SOff[l] + idx];
    const int x = idx % W, y = idx / W;
    const float ax = (x + 0.5f) * st, ay = (y + 0.5f) * st;
    const float dl = fmaxf(rg[0 * P + idx] * sc, 0.f) * st;
    const float dt = fmaxf(rg[1 * P + idx] * sc, 0.f) * st;
    const float dr = fmaxf(rg[2 * P + idx] * sc, 0.f) * st;
    const float db = fmaxf(rg[3 * P + idx] * sc, 0.f) * st;
    const float bx1 = ax - dl, by1 = ay - dt, bx2 = ax + dr, by2 = ay + db;
    catScore[s] = __uint_as_float(kb);
    catLabel[s] = lab;
    catBox[s * 4 + 0] = bx1; catBox[s * 4 + 1] = by1;
    catBox[s * 4 + 2] = bx2; catBox[s * 4 + 3] = by2;
    const float off = (float)lab * 100000.0f;
    catBoxOff[s * 4 + 0] = bx1 + off; catBoxOff[s * 4 + 1] = by1 + off;
    catBoxOff[s * 4 + 2] = bx2 + off; catBoxOff[s * 4 + 3] = by2 + off;
}

// ---------------------------------------------------------------------------
// K4: global argsort(-where(valid,score,-inf)) via 8192-wide bitonic in LDS
// ---------------------------------------------------------------------------
__global__ void sort_cat_kernel(const float* __restrict__ catScore,
                                int* __restrict__ order) {
    extern __shared__ unsigned long long scomb[];   // 8192 * 8B = 64KB dynamic LDS
    const int n = 8192;
    for (int i = threadIdx.x; i < n; i += blockDim.x) {
        unsigned long long key;
        if (i < NCAT) {
            float s = catScore[i];
            unsigned sb = (s >= CONF_TH) ? __float_as_uint(s) : 0u;   // invalid -> -inf class
            key = ((unsigned long long)sb << 32) | (unsigned)(~i);    // stable: idx ascending
        } else {
            key = (unsigned long long)(unsigned)(~i);                 // padding sorts after real
        }
        scomb[i] = key;
    }
    bitonic_desc(scomb, n);
    for (int r = threadIdx.x; r < NCAT; r += blockDim.x)
        order[r] = (int)(~(unsigned)scomb[r]);
}

// ---------------------------------------------------------------------------
// K5: pairwise suppression bitmask (sorted order, j > i only)
// ---------------------------------------------------------------------------
__global__ void mask_kernel(const float* __restrict__ bo,
                            const int* __restrict__ order,
                            unsigned long long* __restrict__ mask) {
    const int t = blockIdx.x * blockDim.x + threadIdx.x;
    const int total = NCAT * NBLK;
    if (t >= total) return;
    const int i  = t / NBLK;
    const int jb = t % NBLK;
    const int bi = order[i];
    __builtin_prefetch(bo + (size_t)bi * 4, 0, 0);
    const float x1 = bo[bi * 4 + 0], y1 = bo[bi * 4 + 1];
    const float x2 = bo[bi * 4 + 2], y2 = bo[bi * 4 + 3];
    const float ai = (x2 - x1) * (y2 - y1);
    unsigned long long bits = 0;
    const int j0 = jb * 64;
    for (int b = 0; b < 64; ++b) {
        const int j = j0 + b;
        if (j < NCAT && j > i) {
            const int bj = order[j];
            const float X1 = bo[bj * 4 + 0], Y1 = bo[bj * 4 + 1];
            const float X2 = bo[bj * 4 + 2], Y2 = bo[bj * 4 + 3];
            const float aj  = (X2 - X1) * (Y2 - Y1);
            const float xx1 = fmaxf(x1, X1), yy1 = fmaxf(y1, Y1);
            const float xx2 = fminf(x2, X2), yy2 = fminf(y2, Y2);
            const float inter = fmaxf(1e-10f, xx2 - xx1) * fmaxf(1e-10f, yy2 - yy1);
            const float ovr   = inter / (ai + aj - inter + 1e-14f);
            if (ovr > NMS_TH) bits |= (1ull << b);
        }
    }
    mask[(size_t)i * NBLK + jb] = bits;
}

// ---------------------------------------------------------------------------
// K6: serial greedy NMS resolve. ONE wave32: 67 keep-words live in registers
//     (3 per lane), mask rows double-buffered into LDS via async loads.
// ---------------------------------------------------------------------------
__device__ inline void nms_load_chunk(const unsigned long long* __restrict__ mask,
                                      unsigned long long* buf, int ch, int lane) {
    const int    CH   = 64;
    const int    WPC  = CH * NBLK;                 // 4288 words / chunk
    const size_t base = (size_t)ch * WPC;
    const size_t tot  = (size_t)NCAT * NBLK;
    for (int k = lane; k < WPC; k += 32) {
        const size_t g = base + k;
        if (g < tot) {
#ifdef GFX1250_ASYNC
            unsigned           la = (unsigned)(size_t)&buf[k];
            unsigned long long ga = (unsigned long long)(size_t)(mask + g);
            asm volatile("global_load_async_to_lds_b64 %0, %1, off" :: "v"(la), "v"(ga) : "memory");
#else
            buf[k] = mask[g];
#endif
        }
    }
}

__global__ void nms_kernel(const unsigned long long* __restrict__ mask,
                           const float* __restrict__ catScore,
                           const int* __restrict__ order,
                           int* __restrict__ keepOut) {
    extern __shared__ unsigned long long chbuf[];  // 2 * 64 * 67 u64 = 68608 B
    const int CH  = 64;
    const int WPC = CH * NBLK;
    const int lane = threadIdx.x;                  // blockDim == 32 (one wave)

    // keep bits (sorted order) in registers: lane owns words lane, lane+32, lane+64
    unsigned long long w0 = 0, w1 = 0, w2 = 0;
    for (int wr = 0; wr < 3; ++wr) {
        const int wi = lane + wr * 32;
        if (wi < NBLK) {
            unsigned long long w = 0;
            for (int b = 0; b < 64; ++b) {
                const int r = wi * 64 + b;
                if (r < NCAT && catScore[order[r]] >= CONF_TH) w |= (1ull << b);
            }
            if (wr == 0) w0 = w; else if (wr == 1) w1 = w; else w2 = w;
        }
    }

    const int NCH = (NCAT + CH - 1) / CH;          // 67 chunks
    nms_load_chunk(mask, chbuf, 0, lane);
#ifdef GFX1250_ASYNC
    asm volatile("s_wait_asynccnt 0" ::: "memory");
#endif
    for (int ch = 0; ch < NCH; ++ch) {
        if (ch + 1 < NCH)
            nms_load_chunk(mask, chbuf + (size_t)((ch + 1) & 1) * WPC, ch + 1, lane);
        const unsigned long long* rowbase = chbuf + (size_t)(ch & 1) * WPC;
        const int rows = (NCAT - ch * CH < CH) ? (NCAT - ch * CH) : CH;
        for (int r = 0; r < rows; ++r) {
            const int i  = ch * CH + r;
            const int wi = i >> 6;
            const int src = wi & 31;
            const int rg  = wi >> 5;                           // uniform
            unsigned long long cand = (rg == 0) ? w0 : ((rg == 1) ? w1 : w2);
            unsigned long long ww = __shfl(cand, src, 32);     // broadcast word holding bit i
            if ((ww >> (i & 63)) & 1ull) {                     // box i kept -> apply its row
                const unsigned long long* row = rowbase + (size_t)r * NBLK;
                if (lane < NBLK)      w0 &= ~row[lane];
                if (lane + 32 < NBLK) w1 &= ~row[lane + 32];
                if (lane + 64 < NBLK) w2 &= ~row[lane + 64];
            }
        }
#ifdef GFX1250_ASYNC
        asm volatile("s_wait_asynccnt 0" ::: "memory");        // next chunk landed in LDS
#endif
    }

    // scatter keep back to pre-sort (concatenated) order
    for (int wr = 0; wr < 3; ++wr) {
        const int wi = lane + wr * 32;
        if (wi < NBLK) {
            const unsigned long long w = (wr == 0) ? w0 : ((wr == 1) ? w1 : w2);
            for (int b = 0; b < 64; ++b) {
                const int r = wi * 64 + b;
                if (r < NCAT) keepOut[order[r]] = (int)((w >> b) & 1ull);
            }
        }
    }
}

// ---------------------------------------------------------------------------
// K7: finalize -> d_out = [bboxes(4256*4) | scores | labels | keep] as float
// ---------------------------------------------------------------------------
__global__ void finalize_kernel(const float* __restrict__ catBox,
                                const float* __restrict__ catScore,
                                const int* __restrict__ catLabel,
                                const int* __restrict__ keepOrig,
                                float* __restrict__ out) {
    const int s = blockIdx.x * blockDim.x + threadIdx.x;
    if (s >= NCAT) return;
    const float inv = 1.0f / 2048.0f;
    for (int k = 0; k < 4; ++k) {
        float v = catBox[s * 4 + k] * inv;
        out[s * 4 + k] = fminf(fmaxf(v, 0.0f), 1.0f);
    }
    out[NCAT * 4 + s] = catScore[s];
    out[NCAT * 5 + s] = (float)catLabel[s];
    out[NCAT * 6 + s] = keepOrig[s] ? 1.0f : 0.0f;
}

// ---------------------------------------------------------------------------
// launcher
// ---------------------------------------------------------------------------
extern "C" void kernel_launch(void* const* d_in, const int* in_sizes, int n_in,
                              void* d_out, int out_size, void* d_ws, size_t ws_size,
                              hipStream_t stream) {
    (void)in_sizes; (void)n_in; (void)out_size; (void)ws_size;

    const float* cls[NLEV]; const float* reg[NLEV]; const float* ctn[NLEV];
    for (int l = 0; l < NLEV; ++l) {
        cls[l] = (const float*)d_in[3 * l + 0];
        reg[l] = (const float*)d_in[3 * l + 1];
        ctn[l] = (const float*)d_in[3 * l + 2];
    }
    const float* scales = (const float*)d_in[15];

    // workspace layout (256B aligned blocks)
    char* ws = (char*)d_ws;
    size_t off = 0;
    auto alloc = [&](size_t bytes) -> size_t {
        size_t o = off; off += (bytes + 255) & ~(size_t)255; return o;
    };
    const size_t o_keys = alloc((size_t)TOT_POS * 4);
    const size_t o_lab  = alloc((size_t)TOT_POS * 4);
    const size_t o_sel  = alloc((size_t)NCAT * 4);
    const size_t o_cs   = alloc((size_t)NCAT * 4);
    const size_t o_cl   = alloc((size_t)NCAT * 4);
    const size_t o_cb   = alloc((size_t)NCAT * 16);
    const size_t o_cbo  = alloc((size_t)NCAT * 16);
    const size_t o_ord  = alloc((size_t)NCAT * 4);
    const size_t o_keep = alloc((size_t)NCAT * 4);
    const size_t o_mask = alloc((size_t)NCAT * NBLK * 8);

    unsigned*           keys    = (unsigned*)(ws + o_keys);
    int*                labels  = (int*)(ws + o_lab);
    int*                sel     = (int*)(ws + o_sel);
    float*              catS    = (float*)(ws + o_cs);
    int*                catL    = (int*)(ws + o_cl);
    float*              catB    = (float*)(ws + o_cb);
    float*              catBO   = (float*)(ws + o_cbo);
    int*                order   = (int*)(ws + o_ord);
    int*                keepO   = (int*)(ws + o_keep);
    unsigned long long* maskBuf = (unsigned long long*)(ws + o_mask);

    const int P[NLEV]      = {65536, 16384, 4096, 1024, 256};
    const int posOff[NLEV] = {0, 65536, 81920, 86016, 87040};
    const int catOff[NLEV] = {0, 1000, 2000, 3000, 4000};

    for (int l = 0; l < NLEV; ++l)
        decode_kernel<<<P[l] / 256, 256, 0, stream>>>(cls[l], ctn[l],
                                                      keys + posOff[l], labels + posOff[l], P[l]);
    for (int l = 0; l < NLEV; ++l)
        topk_kernel<<<1, 1024, 0, stream>>>(keys + posOff[l], P[l], sel + catOff[l]);

    gather_kernel<<<(NCAT + 255) / 256, 256, 0, stream>>>(sel, keys, labels,
        reg[0], reg[1], reg[2], reg[3], reg[4], scales, catS, catL, catB, catBO);

    sort_cat_kernel<<<1, 1024, 8192 * sizeof(unsigned long long), stream>>>(catS, order);

    const int totalMask = NCAT * NBLK;
    mask_kernel<<<(totalMask + 255) / 256, 256, 0, stream>>>(catBO, order, maskBuf);

    nms_kernel<<<1, 32, 2 * 64 * NBLK * sizeof(unsigned long long), stream>>>(
        maskBuf, catS, order, keepO);

    finalize_kernel<<<(NCAT + 255) / 256, 256, 0, stream>>>(catB, catS, catL, keepO,
                                                            (float*)d_out);
}